// Net_5446018531619
// MI455X (gfx1250) — compile-verified
//
#include <hip/hip_runtime.h>
#include <math.h>

typedef float v2f __attribute__((ext_vector_type(2)));
typedef float v8f __attribute__((ext_vector_type(8)));

#define F_IN  128
#define F_HID 32
#define F_OUT 7

// Native CDNA5 non-returning float atomic-add, device scope.
// Non-returning (no vdst, no th modifier) -> tracked on STOREcnt; S_ENDPGM's
// implicit wait-idle drains it before the kernel is considered complete.
__device__ __forceinline__ void gfadd_dev(float* p, float v) {
    asm volatile("global_atomic_add_f32 %0, %1, off scope:SCOPE_DEV"
                 :: "v"(p), "v"(v)
                 : "memory");
}

// ---------------- degree / normalization ----------------
__global__ void k_init_deg(float* __restrict__ deg, int n) {
    int i = blockIdx.x * blockDim.x + threadIdx.x;
    if (i < n) deg[i] = 1.0f;  // self loop contributes 1
}

__global__ void k_deg_accum(const long long* __restrict__ dst,
                            float* __restrict__ deg, long long e) {
    long long i = (long long)blockIdx.x * blockDim.x + threadIdx.x;
    if (i < e) gfadd_dev(&deg[(int)dst[i]], 1.0f);
}

__global__ void k_dinv(float* __restrict__ deg_dinv, int n) {
    int i = blockIdx.x * blockDim.x + threadIdx.x;
    if (i < n) deg_dinv[i] = rsqrtf(deg_dinv[i]);  // deg >= 1 always
}

// ---------------- GEMM1: h1 = x @ W1  (f32 WMMA 16x16x4) ----------------
// One wave computes a 16-row x 32-col output tile (two 16x16 N-tiles).
__global__ __launch_bounds__(256)
void k_gemm1_wmma(const float* __restrict__ x, const float* __restrict__ W1,
                  float* __restrict__ h1, int n) {
    int wave = (int)((blockIdx.x * blockDim.x + threadIdx.x) >> 5);
    int lane = threadIdx.x & 31;
    int rowBase = wave * 16;
    if (rowBase >= n) return;                 // wave-uniform exit

    int lane15 = lane & 15;                   // A: M row / B: N col
    int khalf  = (lane >> 4) << 1;            // 0 or 2 (K sub-offset per half-wave)
    int row = rowBase + lane15; if (row >= n) row = n - 1;   // clamped load
    const float* xrow = x + (size_t)row * F_IN;

    v8f c0 = {}; v8f c1 = {};
    #pragma unroll
    for (int k = 0; k < F_IN; k += 4) {
        int kk = k + khalf;
        v2f a  = { xrow[kk], xrow[kk + 1] };
        v2f b0 = { W1[kk * F_HID + lane15],      W1[(kk + 1) * F_HID + lane15] };
        v2f b1 = { W1[kk * F_HID + 16 + lane15], W1[(kk + 1) * F_HID + 16 + lane15] };
        c0 = __builtin_amdgcn_wmma_f32_16x16x4_f32(false, a, false, b0, (short)0, c0, false, false);
        c1 = __builtin_amdgcn_wmma_f32_16x16x4_f32(false, a, false, b1, (short)0, c1, false, false);
    }

    int rtop = (lane >> 4) * 8;               // C/D: lanes 16-31 hold M = v+8
    #pragma unroll
    for (int v = 0; v < 8; ++v) {
        int r = rowBase + rtop + v;
        if (r < n) {
            h1[(size_t)r * F_HID + lane15]      = c0[v];
            h1[(size_t)r * F_HID + 16 + lane15] = c1[v];
        }
    }
}

// ---------------- scatter layer 1: acc1[dst] += h1[src] * norm ----------------
// One wave per edge, lane = feature (32 features) -> coalesced f32 atomics.
__global__ void k_scatter1(const long long* __restrict__ src,
                           const long long* __restrict__ dst,
                           const float* __restrict__ dinv,
                           const float* __restrict__ h1,
                           float* __restrict__ acc1, long long e) {
    long long idx  = (long long)blockIdx.x * blockDim.x + threadIdx.x;
    long long eidx = idx >> 5;
    int f = (int)(idx & 31);
    if (eidx >= e) return;
    int s = (int)src[eidx];
    int d = (int)dst[eidx];
    float norm = dinv[s] * dinv[d];
    gfadd_dev(&acc1[(size_t)d * F_HID + f], h1[(size_t)s * F_HID + f] * norm);
}

// ---------------- finalize layer 1: self-loop + bias + ReLU (in place) ----------------
__global__ void k_finalize1(const float* __restrict__ h1, const float* __restrict__ dinv,
                            const float* __restrict__ b1, float* __restrict__ acc1_h2, int n) {
    long long idx = (long long)blockIdx.x * blockDim.x + threadIdx.x;
    int node = (int)(idx >> 5);
    int f    = (int)(idx & 31);
    if (node >= n) return;
    float di = dinv[node];
    size_t o = (size_t)node * F_HID + f;
    float v = acc1_h2[o] + h1[o] * di * di + b1[f];
    acc1_h2[o] = fmaxf(v, 0.0f);
}

// ---------------- GEMM2: h3 = h2 @ W2  (f32 WMMA, N padded 7->16) ----------------
__global__ __launch_bounds__(256)
void k_gemm2_wmma(const float* __restrict__ h2, const float* __restrict__ W2,
                  float* __restrict__ h3, int n) {
    int wave = (int)((blockIdx.x * blockDim.x + threadIdx.x) >> 5);
    int lane = threadIdx.x & 31;
    int rowBase = wave * 16;
    if (rowBase >= n) return;

    int lane15 = lane & 15;
    int khalf  = (lane >> 4) << 1;
    int row = rowBase + lane15; if (row >= n) row = n - 1;
    const float* hrow = h2 + (size_t)row * F_HID;
    bool colOK = (lane15 < F_OUT);

    v8f c0 = {};
    #pragma unroll
    for (int k = 0; k < F_HID; k += 4) {
        int kk = k + khalf;
        v2f a = { hrow[kk], hrow[kk + 1] };
        v2f b = { colOK ? W2[kk * F_OUT + lane15] : 0.0f,
                  colOK ? W2[(kk + 1) * F_OUT + lane15] : 0.0f };
        c0 = __builtin_amdgcn_wmma_f32_16x16x4_f32(false, a, false, b, (short)0, c0, false, false);
    }

    int rtop = (lane >> 4) * 8;
    #pragma unroll
    for (int v = 0; v < 8; ++v) {
        int r = rowBase + rtop + v;
        if (r < n && colOK) h3[(size_t)r * F_OUT + lane15] = c0[v];
    }
}

// ---------------- scatter layer 2: acc2[dst] += h3[src] * norm ----------------
__global__ void k_scatter2(const long long* __restrict__ src,
                           const long long* __restrict__ dst,
                           const float* __restrict__ dinv,
                           const float* __restrict__ h3,
                           float* __restrict__ acc2, long long e) {
    long long eidx = (long long)blockIdx.x * blockDim.x + threadIdx.x;
    if (eidx >= e) return;
    int s = (int)src[eidx];
    int d = (int)dst[eidx];
    float norm = dinv[s] * dinv[d];
    #pragma unroll
    for (int c = 0; c < F_OUT; ++c)
        gfadd_dev(&acc2[(size_t)d * F_OUT + c], h3[(size_t)s * F_OUT + c] * norm);
}

// ---------------- finalize layer 2: self-loop + bias + log_softmax ----------------
__global__ void k_finalize2(const float* __restrict__ h3, const float* __restrict__ dinv,
                            const float* __restrict__ b2, const float* __restrict__ acc2,
                            float* __restrict__ out, int n) {
    int i = blockIdx.x * blockDim.x + threadIdx.x;
    if (i >= n) return;
    float di = dinv[i];
    float d2 = di * di;
    float v[F_OUT];
    float m = -INFINITY;
    #pragma unroll
    for (int c = 0; c < F_OUT; ++c) {
        size_t o = (size_t)i * F_OUT + c;
        v[c] = acc2[o] + h3[o] * d2 + b2[c];
        m = fmaxf(m, v[c]);
    }
    float s = 0.0f;
    #pragma unroll
    for (int c = 0; c < F_OUT; ++c) s += expf(v[c] - m);
    float lse = logf(s) + m;
    #pragma unroll
    for (int c = 0; c < F_OUT; ++c) out[(size_t)i * F_OUT + c] = v[c] - lse;
}

// ---------------- launch ----------------
extern "C" void kernel_launch(void* const* d_in, const int* in_sizes, int n_in,
                              void* d_out, int out_size, void* d_ws, size_t ws_size,
                              hipStream_t stream) {
    const float*     x  = (const float*)d_in[0];
    const long long* ei = (const long long*)d_in[1];   // int64 [2, E]
    const float*     W1 = (const float*)d_in[2];
    const float*     b1 = (const float*)d_in[3];
    const float*     W2 = (const float*)d_in[4];
    const float*     b2 = (const float*)d_in[5];
    float* out = (float*)d_out;

    int n = in_sizes[0] / F_IN;
    long long e = in_sizes[1] / 2;
    const long long* src = ei;        // row 0
    const long long* dst = ei + e;    // row 1

    // workspace layout (floats): dinv | h1(32n) | acc1->h2(32n) | h3(7n) | acc2(7n)
    float* ws   = (float*)d_ws;
    float* dinv = ws;
    float* h1   = dinv + n;
    float* acc1 = h1 + (size_t)F_HID * n;
    float* h3   = acc1 + (size_t)F_HID * n;
    float* acc2 = h3 + (size_t)F_OUT * n;

    hipMemsetAsync(acc1, 0, (size_t)F_HID * n * sizeof(float), stream);
    hipMemsetAsync(acc2, 0, (size_t)F_OUT * n * sizeof(float), stream);

    const int B = 256;
    k_init_deg<<<(n + B - 1) / B, B, 0, stream>>>(dinv, n);
    k_deg_accum<<<(unsigned)((e + B - 1) / B), B, 0, stream>>>(dst, dinv, e);
    k_dinv<<<(n + B - 1) / B, B, 0, stream>>>(dinv, n);

    long long waves = (n + 15) / 16;                 // one wave per 16 output rows
    unsigned gGemm = (unsigned)((waves * 32 + B - 1) / B);
    k_gemm1_wmma<<<gGemm, B, 0, stream>>>(x, W1, h1, n);

    long long t1 = e * 32;                           // wave per edge, lane per feature
    k_scatter1<<<(unsigned)((t1 + B - 1) / B), B, 0, stream>>>(src, dst, dinv, h1, acc1, e);

    long long t2 = (long long)n * 32;
    k_finalize1<<<(unsigned)((t2 + B - 1) / B), B, 0, stream>>>(h1, dinv, b1, acc1, n);

    k_gemm2_wmma<<<gGemm, B, 0, stream>>>(acc1, W2, h3, n);

    k_scatter2<<<(unsigned)((e + B - 1) / B), B, 0, stream>>>(src, dst, dinv, h3, acc2, e);

    k_finalize2<<<(n + B - 1) / B, B, 0, stream>>>(h3, dinv, b2, acc2, out, n);

    (void)n_in; (void)out_size; (void)ws_size;
}